// XaiYoloWrapper_56478819942463
// MI455X (gfx1250) — compile-verified
//
#include <hip/hip_runtime.h>
#include <cstdint>
#include <cstddef>

#define NCLS   80
#define ROW    85
#define NANCH  25200
#define SORTN  32768          // next pow2 >= NANCH, 256 KB of u64 keys in LDS (CDNA5: 320 KB/WG)
#define KNMS   2048
#define CONF_T 0.25f
#define IOU_T  0.45f
#define MAXDET 1000
#define MAXWH  7680.0f

typedef float v2f __attribute__((ext_vector_type(2)));
typedef float v8f __attribute__((ext_vector_type(8)));

// ---------------------------------------------------------------------------
// Kernel 1: per-image score pass + in-LDS bitonic top-K.
// Key = (f32 bits of best-score << 32) | (0xFFFFFFFF - anchor_index); invalid -> 0.
// Descending sort on this u64 == lax.top_k ordering (value desc, index asc on ties).
// ---------------------------------------------------------------------------
__global__ __launch_bounds__(1024)
void yolo_score_sort(const float* __restrict__ pred,
                     unsigned long long* __restrict__ topkeys)
{
    extern __shared__ unsigned long long keys[];   // SORTN * 8 = 256 KB
    const int b = blockIdx.x;
    const int t = threadIdx.x;
    const float* img = pred + (size_t)b * NANCH * ROW;

    // ---- scoring: obj-gated (only ~4% of anchors need the 80 class reads) ----
    for (int n = t; n < SORTN; n += 1024) {
        unsigned long long key = 0ull;
        if (n < NANCH) {
            const float* row = img + (size_t)n * ROW;
            float obj = row[4];
            if (obj > CONF_T) {
                __builtin_prefetch(row + 5, 0, 1);   // global_prefetch_b8
                float best = -1.0f;
                #pragma unroll 8
                for (int c = 0; c < NCLS; ++c) {
                    float s = row[5 + c] * obj;
                    best = fmaxf(best, s);
                }
                if (best > CONF_T) {
                    key = ((unsigned long long)__float_as_uint(best) << 32)
                        | (unsigned long long)(0xFFFFFFFFu - (unsigned)n);
                }
            }
        }
        keys[n] = key;
    }
    __syncthreads();

    // ---- bitonic sort, descending ----
    for (int k = 2; k <= SORTN; k <<= 1) {
        for (int j = k >> 1; j > 0; j >>= 1) {
            for (int e = t; e < SORTN; e += 1024) {
                int p = e ^ j;
                if (p > e) {
                    bool up = ((e & k) == 0);
                    unsigned long long a = keys[e], c = keys[p];
                    if ((a < c) == up) { keys[e] = c; keys[p] = a; }
                }
            }
            __syncthreads();
        }
    }

    for (int k = t; k < KNMS; k += 1024)
        topkeys[(size_t)b * KNMS + k] = keys[k];
}

// ---------------------------------------------------------------------------
// Kernel 2: per-image box decode + blocked bitmask NMS + WMMA masked GEMV.
// ---------------------------------------------------------------------------
__global__ __launch_bounds__(1024)
void yolo_nms_reduce(const float* __restrict__ pred,
                     const unsigned long long* __restrict__ topkeys,
                     float* __restrict__ out)
{
    extern __shared__ char smem_raw[];
    float*    bx1   = (float*)smem_raw;            // [KNMS]
    float*    by1   = bx1 + KNMS;
    float*    bx2   = by1 + KNMS;
    float*    by2   = bx2 + KNMS;
    float*    barea = by2 + KNMS;
    float*    wgt   = barea + KNMS;                // obj weight, zeroed if not final
    unsigned* sidx  = (unsigned*)(wgt + KNMS);     // anchor index per slot
    unsigned* keepm = sidx + KNMS;                 // [64] keep bitmask (2048 bits)
    unsigned* supm  = keepm + 64;                  // [32][64] suppression masks
    int*      wbase = (int*)(supm + 32 * 64);      // [64] prefix of popcounts
    int*      pc    = wbase + 64;                  // [64]

    const int b    = blockIdx.x;
    const int t    = threadIdx.x;
    const int lane = t & 31;
    const int wv   = t >> 5;                       // wave id, 0..31
    const float* img = pred + (size_t)b * NANCH * ROW;

    if (t < 64) keepm[t] = 0u;
    __syncthreads();

    // ---- gather top-K anchors, decode class-offset boxes ----
    for (int k = t; k < KNMS; k += 1024) {
        unsigned long long key = topkeys[(size_t)b * KNMS + k];
        unsigned sb = (unsigned)(key >> 32);
        float x1 = 0, y1 = 0, x2 = 0, y2 = 0, ar = 0, w = 0;
        unsigned idx = 0;
        if (sb != 0u) {                                     // vtop
            idx = 0xFFFFFFFFu - (unsigned)(key & 0xFFFFFFFFull);
            const float* row = img + (size_t)idx * ROW;
            float cx = row[0], cy = row[1], bw = row[2], bh = row[3];
            float obj = row[4];
            float best = -1.0f; int jc = 0;
            for (int c = 0; c < NCLS; ++c) {
                float s = row[5 + c] * obj;
                if (s > best) { best = s; jc = c; }          // first-max, as argmax
            }
            float off = (float)jc * MAXWH;
            x1 = cx - bw * 0.5f + off;  y1 = cy - bh * 0.5f + off;
            x2 = cx + bw * 0.5f + off;  y2 = cy + bh * 0.5f + off;
            ar = (x2 - x1) * (y2 - y1);
            w  = obj;
            atomicOr(&keepm[k >> 5], 1u << (k & 31));        // keep init = vtop
        }
        bx1[k] = x1; by1[k] = y1; bx2[k] = x2; by2[k] = y2;
        barea[k] = ar; wgt[k] = w; sidx[k] = idx;
    }
    __syncthreads();

    // ---- blocked greedy NMS: 64 blocks of 32 i's ----
    for (int blk = 0; blk < KNMS / 32; ++blk) {
        const int i = blk * 32 + wv;                 // each wave owns one i-row
        float ix1 = bx1[i], iy1 = by1[i], ix2 = bx2[i], iy2 = by2[i], iar = barea[i];
        for (int jw = 0; jw < 64; ++jw) {
            int j = jw * 32 + lane;
            float xx1 = fmaxf(ix1, bx1[j]);
            float yy1 = fmaxf(iy1, by1[j]);
            float xx2 = fminf(ix2, bx2[j]);
            float yy2 = fminf(iy2, by2[j]);
            float iw2 = fmaxf(xx2 - xx1, 0.0f);
            float ih2 = fmaxf(yy2 - yy1, 0.0f);
            float inter = iw2 * ih2;
            float uni   = iar + barea[j] - inter;
            float iou   = (uni > 0.0f) ? inter / uni : 0.0f;
            bool  sup   = (iou > IOU_T) && (j > i);
            unsigned m = (unsigned)__ballot(sup);    // wave32 ballot
            if (lane == 0) supm[wv * 64 + jw] = m;
        }
        __syncthreads();
        if (t < 32) {                                // wave 0 resolves serially
            volatile unsigned* vk = keepm;
            for (int il = 0; il < 32; ++il) {
                int ii = blk * 32 + il;
                unsigned kw = vk[ii >> 5];
                if ((kw >> (ii & 31)) & 1u) {        // keep[i] & vtop[i]
                    vk[t]      &= ~supm[il * 64 + t];
                    vk[t + 32] &= ~supm[il * 64 + t + 32];
                }
            }
        }
        __syncthreads();
    }

    // ---- rank = cumsum(keep) - 1 ; final = keep & rank < MAXDET ----
    if (t < 64) pc[t] = __popc(keepm[t]);
    __syncthreads();
    if (t == 0) { int s = 0; for (int w2 = 0; w2 < 64; ++w2) { wbase[w2] = s; s += pc[w2]; } }
    __syncthreads();
    for (int k = t; k < KNMS; k += 1024) {
        unsigned word = keepm[k >> 5];
        int bit = k & 31;
        bool kp = (word >> bit) & 1u;
        int rank = wbase[k >> 5] + __popc(word & ((1u << bit) - 1u));
        if (!(kp && rank < MAXDET)) wgt[k] = 0.0f;
    }
    __syncthreads();

    // ---- masked GEMV out[c] = sum_k wgt[k] * cls[idx_k, c] via f32 WMMA ----
    // Pad M: A[m,kk] = w[kk] broadcast over all 16 rows -> every D row equals out.
    // 5 waves, one 16-class group each; K chunks of 4 accumulated in C.
    if (t < 5 * 32) {
        const int cbase = wv * 16;
        const int half  = lane >> 4;                 // A/B K-split across lane halves
        const int n     = lane & 15;
        v8f acc = {};
        for (int ch = 0; ch < KNMS / 4; ++ch) {
            int k0 = ch * 4;
            float w0 = wgt[k0 + 0], w1 = wgt[k0 + 1];
            float w2 = wgt[k0 + 2], w3 = wgt[k0 + 3];
            if ((w0 == 0.0f) & (w1 == 0.0f) & (w2 == 0.0f) & (w3 == 0.0f))
                continue;                            // wave-uniform skip
            float wA = half ? w2 : w0;
            float wB = half ? w3 : w1;
            int   kA = k0 + (half ? 2 : 0);
            v2f a; a.x = wA; a.y = wB;               // A: lanes0-15 K=0,1 ; lanes16-31 K=2,3
            float b0 = 0.0f, b1 = 0.0f;
            if (wA != 0.0f) b0 = img[(size_t)sidx[kA]     * ROW + 5 + cbase + n];
            if (wB != 0.0f) b1 = img[(size_t)sidx[kA + 1] * ROW + 5 + cbase + n];
            v2f bb; bb.x = b0; bb.y = b1;            // B: same K split, N = lane&15
            acc = __builtin_amdgcn_wmma_f32_16x16x4_f32(
                      false, a, false, bb, (short)0, acc, false, false);
        }
        // D row M=0 lives in acc[0] of lanes 0..15
        if (lane < 16) out[b * NCLS + cbase + lane] = acc[0];
    }
}

// ---------------------------------------------------------------------------
extern "C" void kernel_launch(void* const* d_in, const int* in_sizes, int n_in,
                              void* d_out, int out_size, void* d_ws, size_t ws_size,
                              hipStream_t stream)
{
    const float* pred = (const float*)d_in[0];          // (16, 25200, 85) f32
    float* out = (float*)d_out;                          // (16, 80) f32
    unsigned long long* topkeys = (unsigned long long*)d_ws;  // 16*2048*8 = 256 KB

    const int B = 16;
    const size_t sort_smem = (size_t)SORTN * 8;          // 256 KB (CDNA5 LDS)
    const size_t nms_smem  = 6 * KNMS * 4                // box coords/area/wgt
                           + KNMS * 4                    // sidx
                           + 64 * 4 + 32 * 64 * 4        // keep + sup masks
                           + 64 * 4 + 64 * 4;            // prefix scratch

    // Allow >48KB dynamic LDS (host-side attribute, not a stream op).
    hipFuncSetAttribute((const void*)yolo_score_sort,
                        hipFuncAttributeMaxDynamicSharedMemorySize, (int)sort_smem);
    hipFuncSetAttribute((const void*)yolo_nms_reduce,
                        hipFuncAttributeMaxDynamicSharedMemorySize, (int)nms_smem);

    yolo_score_sort<<<B, 1024, sort_smem, stream>>>(pred, topkeys);
    yolo_nms_reduce<<<B, 1024, nms_smem, stream>>>(pred, topkeys, out);
}